// ContrastiveLoss_2018634629598
// MI455X (gfx1250) — compile-verified
//
#include <hip/hip_runtime.h>
#include <hip/hip_bf16.h>

typedef float v2f __attribute__((ext_vector_type(2)));
typedef float v8f __attribute__((ext_vector_type(8)));

#define D 256
#define NROWS 4096   // 2B
#define BHALF 2048
#define NTILE 256    // NROWS/16

#define WMMA_F32(a, b, c) \
    __builtin_amdgcn_wmma_f32_16x16x4_f32(false, (a), false, (b), (short)0, (c), false, false)

// ---------------------------------------------------------------------------
// 1) L2-normalize rows of [z_i; z_j] -> zn[4096][256]. One wave per row.
// ---------------------------------------------------------------------------
__global__ void nk_normalize(const float* __restrict__ zi,
                             const float* __restrict__ zj,
                             float* __restrict__ zn) {
    const int wavesPerBlk = blockDim.x >> 5;
    const int row  = blockIdx.x * wavesPerBlk + (threadIdx.x >> 5);
    const int lane = threadIdx.x & 31;
    if (row >= NROWS) return;
    const float* src = (row < BHALF) ? (zi + (size_t)row * D)
                                     : (zj + (size_t)(row - BHALF) * D);
    float x[8];
    float ss = 0.f;
#pragma unroll
    for (int j = 0; j < 8; ++j) {
        x[j] = src[lane + 32 * j];
        ss += x[j] * x[j];
    }
    ss += __shfl_xor(ss, 16, 32);
    ss += __shfl_xor(ss, 8, 32);
    ss += __shfl_xor(ss, 4, 32);
    ss += __shfl_xor(ss, 2, 32);
    ss += __shfl_xor(ss, 1, 32);
    const float scale = 1.0f / fmaxf(sqrtf(ss), 1e-12f);
    float* dst = zn + (size_t)row * D;
#pragma unroll
    for (int j = 0; j < 8; ++j) dst[lane + 32 * j] = x[j] * scale;
}

// ---------------------------------------------------------------------------
// 2) Register-blocked 2x4 WMMA tiles: each wave computes a 32x64 patch of
//    sim = zn*zn^T as 8 x (16x16) accumulators. Per K-step: 6 b64 loads
//    (2 A-frags + 4 B-frags) feed 8 V_WMMA_F32_16X16X4_F32 ops.
//    f32 frag layout (A 16x4 == B 4x16 for Z*Z^T): lane l=lane&15 -> row
//    tile*16+l, half=lane>>4 -> K pair {2h,2h+1}. C: acc[v] = row v+half*8,
//    col lane&15. Row-sums of exp(2*c) -> partial[row][jtile] (no atomics).
// ---------------------------------------------------------------------------
__global__ void nk_simtile(const float* __restrict__ zn,
                           float* __restrict__ partial) {
    const int wavesPerBlk = blockDim.x >> 5;
    const int wave = blockIdx.x * wavesPerBlk + (threadIdx.x >> 5);
    const int ig = wave >> 6;        // 0..127 : 32-row strip
    const int jg = wave & 63;        // 0..63  : 64-col strip
    const int lane = threadIdx.x & 31;
    const int half = lane >> 4;
    const int l    = lane & 15;

    const float* A0 = zn + (size_t)(ig * 32 + l) * D + 2 * half;
    const float* A1 = A0 + 16 * D;
    const float* B0 = zn + (size_t)(jg * 64 + l) * D + 2 * half;
    const float* B1 = B0 + 16 * D;
    const float* B2 = B0 + 32 * D;
    const float* B3 = B0 + 48 * D;

    v8f acc00 = {}, acc01 = {}, acc02 = {}, acc03 = {};
    v8f acc10 = {}, acc11 = {}, acc12 = {}, acc13 = {};

#pragma unroll 8
    for (int k = 0; k < D; k += 4) {
        const v2f a0 = *(const v2f*)(A0 + k);
        const v2f a1 = *(const v2f*)(A1 + k);
        const v2f b0 = *(const v2f*)(B0 + k);
        const v2f b1 = *(const v2f*)(B1 + k);
        const v2f b2 = *(const v2f*)(B2 + k);
        const v2f b3 = *(const v2f*)(B3 + k);
        acc00 = WMMA_F32(a0, b0, acc00);
        acc01 = WMMA_F32(a0, b1, acc01);
        acc02 = WMMA_F32(a0, b2, acc02);
        acc03 = WMMA_F32(a0, b3, acc03);
        acc10 = WMMA_F32(a1, b0, acc10);
        acc11 = WMMA_F32(a1, b1, acc11);
        acc12 = WMMA_F32(a1, b2, acc12);
        acc13 = WMMA_F32(a1, b3, acc13);
    }

    v8f* accs[2][4] = {{&acc00, &acc01, &acc02, &acc03},
                       {&acc10, &acc11, &acc12, &acc13}};
#pragma unroll
    for (int i = 0; i < 2; ++i) {
#pragma unroll
        for (int q = 0; q < 4; ++q) {
            const v8f a = *accs[i][q];
#pragma unroll
            for (int v = 0; v < 8; ++v) {
                float e = __expf(2.0f * a[v]);   // 1/TEMPERATURE = 2
                e += __shfl_xor(e, 1, 16);
                e += __shfl_xor(e, 2, 16);
                e += __shfl_xor(e, 4, 16);
                e += __shfl_xor(e, 8, 16);
                if (l == 0) {
                    const int row   = ig * 32 + i * 16 + v + half * 8;
                    const int jtile = jg * 4 + q;
                    partial[(size_t)row * NTILE + jtile] = e;
                }
            }
        }
    }
}

// ---------------------------------------------------------------------------
// 3) Deterministic reduction of 256 tile-partials per row -> S[row]
// ---------------------------------------------------------------------------
__global__ void nk_rowsum(const float* __restrict__ partial,
                          float* __restrict__ S) {
    const int wavesPerBlk = blockDim.x >> 5;
    const int row  = blockIdx.x * wavesPerBlk + (threadIdx.x >> 5);
    const int lane = threadIdx.x & 31;
    if (row >= NROWS) return;
    float s = 0.f;
#pragma unroll
    for (int j = 0; j < NTILE; j += 32) s += partial[(size_t)row * NTILE + j + lane];
    s += __shfl_xor(s, 16, 32);
    s += __shfl_xor(s, 8, 32);
    s += __shfl_xor(s, 4, 32);
    s += __shfl_xor(s, 2, 32);
    s += __shfl_xor(s, 1, 32);
    if (lane == 0) S[row] = s;
}

// ---------------------------------------------------------------------------
// 4) Positive-pair dot and self dot per row (one wave per row)
// ---------------------------------------------------------------------------
__global__ void nk_posdiag(const float* __restrict__ zn,
                           float* __restrict__ posdot,
                           float* __restrict__ diagdot) {
    const int wavesPerBlk = blockDim.x >> 5;
    const int row  = blockIdx.x * wavesPerBlk + (threadIdx.x >> 5);
    const int lane = threadIdx.x & 31;
    if (row >= NROWS) return;
    const int pair = (row + BHALF) & (NROWS - 1);
    const float* a = zn + (size_t)row * D;
    const float* b = zn + (size_t)pair * D;
    float dp = 0.f, dd = 0.f;
#pragma unroll
    for (int j = 0; j < D; j += 32) {
        const float x = a[j + lane];
        dp += x * b[j + lane];
        dd += x * x;
    }
    dp += __shfl_xor(dp, 16, 32); dd += __shfl_xor(dd, 16, 32);
    dp += __shfl_xor(dp, 8, 32);  dd += __shfl_xor(dd, 8, 32);
    dp += __shfl_xor(dp, 4, 32);  dd += __shfl_xor(dd, 4, 32);
    dp += __shfl_xor(dp, 2, 32);  dd += __shfl_xor(dd, 2, 32);
    dp += __shfl_xor(dp, 1, 32);  dd += __shfl_xor(dd, 1, 32);
    if (lane == 0) { posdot[row] = dp; diagdot[row] = dd; }
}

// ---------------------------------------------------------------------------
// 5) loss = mean_i( log(S_i - exp(2*d_i)) - 2*p_i )
// ---------------------------------------------------------------------------
__global__ void nk_finalize(const float* __restrict__ S,
                            const float* __restrict__ posdot,
                            const float* __restrict__ diagdot,
                            float* __restrict__ out) {
    __shared__ float red[256];
    const int t = threadIdx.x;
    float acc = 0.f;
    for (int r = t; r < NROWS; r += 256) {
        const float neg = S[r] - __expf(2.0f * diagdot[r]);
        acc += logf(neg) - 2.0f * posdot[r];
    }
    red[t] = acc;
    __syncthreads();
    for (int s = 128; s > 0; s >>= 1) {
        if (t < s) red[t] += red[t + s];
        __syncthreads();
    }
    if (t == 0) out[0] = red[0] / (float)NROWS;
}

extern "C" void kernel_launch(void* const* d_in, const int* in_sizes, int n_in,
                              void* d_out, int out_size, void* d_ws, size_t ws_size,
                              hipStream_t stream) {
    const float* zi = (const float*)d_in[0];
    const float* zj = (const float*)d_in[1];
    float* out = (float*)d_out;

    float* zn      = (float*)d_ws;              // 4096*256
    float* partial = zn + (size_t)NROWS * D;    // 4096*256
    float* S       = partial + (size_t)NROWS * NTILE;
    float* posdot  = S + NROWS;
    float* diagdot = posdot + NROWS;

    // 1) normalize: 8 waves/block, 1 wave per row
    nk_normalize<<<NROWS / 8, 256, 0, stream>>>(zi, zj, zn);
    // 2) sim tiles: 128 x 64 = 8192 waves (each a 32x64 patch), 4 waves/block
    nk_simtile<<<(128 * 64) / 4, 128, 0, stream>>>(zn, partial);
    // 3) row sums
    nk_rowsum<<<NROWS / 8, 256, 0, stream>>>(partial, S);
    // 4) positive / diagonal dots
    nk_posdiag<<<NROWS / 8, 256, 0, stream>>>(zn, posdot, diagdot);
    // 5) final scalar
    nk_finalize<<<1, 256, 0, stream>>>(S, posdot, diagdot, out);
}